// GlobalMemoryModule_69028714381958
// MI455X (gfx1250) — compile-verified
//
#include <hip/hip_runtime.h>
#include <hip/hip_bf16.h>
#include <math.h>

typedef __attribute__((ext_vector_type(2))) float v2f;
typedef __attribute__((ext_vector_type(8))) float v8f;

#define B_   16
#define CIN  256
#define D_   128
#define C_   32
#define HW   4096
#define PWS  384           // proj_W row stride (Cin + D)
#define EPSF 1e-8f

// ---------------------------------------------------------------------------
// Kernel A: per-batch statistics.
// One block per batch, 128 threads (one per latent dim d).
// Produces ws_bias[b*256+o] and ws_kld[b] (= kl_gw + kl_cat + kl_std).
// ---------------------------------------------------------------------------
__global__ __launch_bounds__(128) void stats_kernel(
    const float* __restrict__ mu_prior,        // (B,D,C)
    const float* __restrict__ log_sigma_prior, // (B,D,C)
    const float* __restrict__ pi_prior,        // (B,C)
    const float* __restrict__ mu_post,         // (B,D)
    const float* __restrict__ log_sigma_post,  // (B,D)
    const float* __restrict__ noise_cluster,   // (B,D)
    const float* __restrict__ noise_z,         // (B,D)
    const float* __restrict__ proj_W,          // (Cin, Cin+D)
    const float* __restrict__ proj_b,          // (Cin,)
    float* __restrict__ ws_bias,               // (B,Cin)
    float* __restrict__ ws_kld)                // (B,)
{
    const int b = blockIdx.x;
    const int t = threadIdx.x;          // 0..127

    __shared__ float s_mupost[D_], s_lspost[D_], s_zpost[D_], s_sig2[D_], s_zc[D_];
    __shared__ float s_lp[D_], s_klg[D_], s_red[D_];

    // per-d quantities
    const float mu = mu_post[b * D_ + t];
    const float ls = log_sigma_post[b * D_ + t];
    const float e  = expf(ls);
    s_mupost[t] = mu;
    s_lspost[t] = ls;
    s_zpost[t]  = mu + e * noise_cluster[b * D_ + t];
    s_sig2[t]   = e * e;                              // sigma_post^2 = exp(2 ls)
    s_zc[t]     = mu + (e + EPSF) * noise_z[b * D_ + t];
    s_red[t]    = -ls + 0.5f * (e * e + mu * mu) - 0.5f;   // kl_std term
    __syncthreads();

    // reduce kl_std over d
    for (int off = 64; off > 0; off >>= 1) {
        if (t < off) s_red[t] += s_red[t + off];
        __syncthreads();
    }

    // accumulate log_prob_c and kl_g: thread t handles cluster c = t&31,
    // d-range group g = t>>5 (4 groups of 32 d's). Loads are coalesced in c.
    const int c = t & 31;
    const int g = t >> 5;
    const float* mup = mu_prior        + (size_t)b * D_ * C_;
    const float* lsp = log_sigma_prior + (size_t)b * D_ * C_;
    const float HALF_LOG_2PI = 0.9189385332046727f;
    float lp = 0.f, klg = 0.f;
    for (int d = g * 32; d < g * 32 + 32; ++d) {
        const float lspr = lsp[d * C_ + c];
        const float mupr = mup[d * C_ + c];
        const float spr2 = expf(2.f * lspr);          // sigma_prior^2
        const float dz   = s_zpost[d] - mupr;
        lp  += -lspr - HALF_LOG_2PI - dz * dz / (2.f * spr2);
        const float dm   = s_mupost[d] - mupr;
        klg += lspr - s_lspost[d] + (s_sig2[d] + dm * dm) / (2.f * spr2 + EPSF) - 0.5f;
    }
    s_lp[t]  = lp;
    s_klg[t] = klg;
    __syncthreads();

    // wave 0 (32 lanes = 32 clusters, wave32) finishes softmax + KL terms
    if (t < 32) {
        float LP = s_lp[t]  + s_lp[t + 32]  + s_lp[t + 64]  + s_lp[t + 96];
        float KG = s_klg[t] + s_klg[t + 32] + s_klg[t + 64] + s_klg[t + 96];
        float m = LP;
        for (int i = 16; i > 0; i >>= 1) m = fmaxf(m, __shfl_xor(m, i, 32));
        const float ex = expf(LP - m);
        float s = ex;
        for (int i = 16; i > 0; i >>= 1) s += __shfl_xor(s, i, 32);
        const float pi  = ex / s;
        const float pip = pi_prior[b * C_ + t];
        float term = pi * KG + pi * (logf(pi + EPSF) - logf(pip + EPSF));
        for (int i = 16; i > 0; i >>= 1) term += __shfl_xor(term, i, 32);
        if (t == 0) ws_kld[b] = term + s_red[0];
    }
    __syncthreads();

    // bias[b,o] = sum_d z_c[d] * Wz[o,d] + proj_b[o], Wz[o,d] = proj_W[o*384 + 256 + d]
    for (int o = t; o < CIN; o += 128) {
        float acc = proj_b[o];
        const float* wrow = proj_W + (size_t)o * PWS + CIN;
        for (int d = 0; d < D_; ++d) acc += s_zc[d] * wrow[d];
        ws_bias[b * CIN + o] = acc;
    }
}

// ---------------------------------------------------------------------------
// Kernel C: fold 16 per-batch KLD partials into the output scalar.
// ---------------------------------------------------------------------------
__global__ void kld_reduce_kernel(const float* __restrict__ ws_kld,
                                  float* __restrict__ out_scalar)
{
    if (threadIdx.x == 0) {
        float s = 0.f;
        for (int i = 0; i < B_; ++i) s += ws_kld[i];
        *out_scalar = s / (float)B_;
    }
}

// ---------------------------------------------------------------------------
// Kernel B: f_out[b,o,hw] = sum_c Wf[o,c]*f_curr[b,c,hw] + bias[b,o]
// via V_WMMA_F32_16X16X4_F32. Block: 256 threads (8 waves).
// Block tile: M=64 x N=128. Wave tile: M=64 x N=16 (4 accumulators), K=256.
// Wf tile (64x256 f32 = 64KB) staged in LDS once per block.
// ---------------------------------------------------------------------------
__global__ __launch_bounds__(256) void gemm_kernel(
    const float* __restrict__ f_curr,   // (B, Cin, HW)
    const float* __restrict__ proj_W,   // (Cin, 384); Wf = cols [0,256)
    const float* __restrict__ bias,     // (B, Cin)
    float* __restrict__ f_out)          // (B, Cin, HW)
{
    __shared__ float lds_w[64 * 256];   // Wf rows [mb*64, mb*64+64), all 256 cols

    const int nb  = blockIdx.x;         // 0..31  (N tiles of 128)
    const int mb  = blockIdx.y;         // 0..3   (M tiles of 64)
    const int b   = blockIdx.z;         // 0..15
    const int tid = threadIdx.x;

    // stage Wf tile into LDS (float4 per thread-iteration)
    {
        const float* wsrc = proj_W + (size_t)(mb * 64) * PWS;
        for (int i = tid; i < 64 * 64; i += 256) {
            const int row = i >> 6;
            const int c4  = (i & 63) << 2;
            const float4 v = *(const float4*)(wsrc + (size_t)row * PWS + c4);
            *(float4*)(&lds_w[row * 256 + c4]) = v;
        }
    }
    __syncthreads();

    const int wave  = tid >> 5;
    const int lane  = tid & 31;
    const int nloc  = lane & 15;        // N within 16-wide tile
    const int khalf = lane >> 4;        // K-pair select (ISA A/B layout)
    const int ncol  = nb * 128 + wave * 16 + nloc;

    const float* bsrc = f_curr + (size_t)b * CIN * HW + ncol;

    v8f acc[4] = {};                    // 4 M-subtiles of 16

    #pragma unroll 4
    for (int kk = 0; kk < CIN; kk += 4) {
        const int k0 = kk + (khalf << 1);
        // B fragment: 4x16 tile of f_curr (K-major, N across lanes)
        v2f bf;
        bf.x = bsrc[(size_t)k0 * HW];
        bf.y = bsrc[(size_t)k0 * HW + HW];
        #pragma unroll
        for (int tt = 0; tt < 4; ++tt) {
            // A fragment: 16x4 tile of Wf from LDS (lane = M row, 2 K per VGPR pair)
            const v2f a = *(const v2f*)(&lds_w[(tt * 16 + nloc) * 256 + k0]);
            acc[tt] = __builtin_amdgcn_wmma_f32_16x16x4_f32(
                false, a, false, bf, (short)0, acc[tt], false, false);
        }
    }

    // epilogue: add bias, non-temporal store (keep L2 for f_curr reuse)
    const float* biasb = bias + b * CIN;
    float* outp = f_out + (size_t)b * CIN * HW + ncol;
    const int mtop = mb * 64 + (khalf << 3);
    #pragma unroll
    for (int tt = 0; tt < 4; ++tt) {
        #pragma unroll
        for (int r = 0; r < 8; ++r) {
            const int mrow = mtop + tt * 16 + r;
            const float v = acc[tt][r] + biasb[mrow];
            __builtin_nontemporal_store(v, outp + (size_t)mrow * HW);
        }
    }
}

// ---------------------------------------------------------------------------
extern "C" void kernel_launch(void* const* d_in, const int* in_sizes, int n_in,
                              void* d_out, int out_size, void* d_ws, size_t ws_size,
                              hipStream_t stream) {
    const float* f_curr          = (const float*)d_in[0];
    const float* mu_prior        = (const float*)d_in[1];
    const float* log_sigma_prior = (const float*)d_in[2];
    const float* pi_prior        = (const float*)d_in[3];
    const float* mu_post         = (const float*)d_in[4];
    const float* log_sigma_post  = (const float*)d_in[5];
    const float* noise_cluster   = (const float*)d_in[6];
    const float* noise_z         = (const float*)d_in[7];
    const float* proj_W          = (const float*)d_in[8];
    const float* proj_b          = (const float*)d_in[9];

    float* out     = (float*)d_out;                 // f_out flat, then kld scalar
    float* ws_kld  = (float*)d_ws;                  // 16 floats
    float* ws_bias = (float*)d_ws + 16;             // 16*256 floats

    stats_kernel<<<B_, 128, 0, stream>>>(mu_prior, log_sigma_prior, pi_prior,
                                         mu_post, log_sigma_post,
                                         noise_cluster, noise_z,
                                         proj_W, proj_b, ws_bias, ws_kld);

    kld_reduce_kernel<<<1, 32, 0, stream>>>(ws_kld, out + (size_t)B_ * CIN * HW);

    gemm_kernel<<<dim3(32, 4, B_), 256, 0, stream>>>(f_curr, proj_W, ws_bias, out);
}